// SpikeRNNCell_317827580114
// MI455X (gfx1250) — compile-verified
//
#include <hip/hip_runtime.h>

typedef float v2f __attribute__((ext_vector_type(2)));
typedef float v8f __attribute__((ext_vector_type(8)));

#define T_STEPS 16
#define ROWS    4096      // B * L = 32 * 128
#define DIN     512
#define DOUT    512
#define MTILE   16        // M rows per block (one WMMA tile)
#define NTILE   32        // N cols per block (2 waves x 16)

// Fused: spikes[t, r, n] = LIF( x[t, r, :] @ W[n, :] + b[n] )
// One wave computes a 16x16 output tile; K=512 reduced via 128x
// V_WMMA_F32_16X16X4_F32 per timestep; membrane potential v lives in
// 8 accumulator-layout VGPRs across the t loop (never spilled to HBM).
__global__ __launch_bounds__(64) void lif_gemm_fused(
    const float* __restrict__ x,     // [T, ROWS, DIN]
    const float* __restrict__ W,     // [DOUT, DIN] row-major
    const float* __restrict__ bias,  // [DOUT]
    float* __restrict__ out)         // [T, ROWS, DOUT]
{
    __shared__ float wtile[NTILE * DIN];   // 64 KB: W rows for this block

    const int tid  = threadIdx.x;
    const int wave = tid >> 5;             // 0..1
    const int lane = tid & 31;
    const int half = lane >> 4;            // lane 16..31 -> K+2 / M+8 half
    const int lm   = lane & 15;

    const int row_base = blockIdx.x * MTILE;        // 0..4080
    const int n_block  = blockIdx.y * NTILE;        // 0..480
    const int n_base   = n_block + wave * 16;

    // ---- stage W[n_block .. n_block+31][0..511] into LDS (float4 copies) ----
    {
        const float4* wsrc = reinterpret_cast<const float4*>(W + n_block * DIN);
        float4*       wdst = reinterpret_cast<float4*>(wtile);
        #pragma unroll 4
        for (int i = tid; i < (NTILE * DIN) / 4; i += 64)
            wdst[i] = wsrc[i];
    }
    __syncthreads();

    const float bval = bias[n_base + lm];

    // A fragment: lane holds x[row_base+lm][k0 + 2*half + {0,1}]  (b64 load)
    const float* aptr0 = x + (row_base + lm) * DIN + 2 * half;
    // B fragment: lane holds W[n_base+lm][k0 + 2*half + {0,1}] from LDS
    const float* bptr  = wtile + (wave * 16 + lm) * DIN + 2 * half;

    // Membrane potential in C/D-matrix layout: VGPR g = row (g + 8*half), col lm
    v8f v = {};

    float* outp = out + (row_base + 8 * half) * DOUT + n_base + lm;

    for (int t = 0; t < T_STEPS; ++t) {
        // h = x_t @ W^T + b : start accumulator at bias (same col per lane)
        v8f acc = { bval, bval, bval, bval, bval, bval, bval, bval };

        const float* ap = aptr0 + t * (ROWS * DIN);
        #pragma unroll 4
        for (int j = 0; j < DIN / 4; ++j) {
            v2f a = *reinterpret_cast<const v2f*>(ap + 4 * j);
            v2f bf = *reinterpret_cast<const v2f*>(bptr + 4 * j);
            acc = __builtin_amdgcn_wmma_f32_16x16x4_f32(
                false, a, false, bf, (short)0, acc,
                /*reuse_a=*/false, /*reuse_b=*/false);
        }

        // LIF: v = v + (h - v)/tau (tau=2, v_reset=0); spike at v>=1; hard reset
        float* op = outp + t * (ROWS * DOUT);
        #pragma unroll
        for (int g = 0; g < 8; ++g) {
            float vv = v[g] + (acc[g] - v[g]) * 0.5f;
            bool  s  = (vv >= 1.0f);
            op[g * DOUT] = s ? 1.0f : 0.0f;
            v[g] = s ? 0.0f : vv;
        }
    }
}

extern "C" void kernel_launch(void* const* d_in, const int* in_sizes, int n_in,
                              void* d_out, int out_size, void* d_ws, size_t ws_size,
                              hipStream_t stream) {
    const float* x    = (const float*)d_in[0];  // [16, 32, 128, 512]
    const float* W    = (const float*)d_in[1];  // [512, 512]
    const float* bias = (const float*)d_in[2];  // [512]
    float* out = (float*)d_out;                 // [16, 32, 128, 512]

    dim3 grid(ROWS / MTILE, DOUT / NTILE);      // 256 x 16
    dim3 block(64);                             // 2 waves (wave32)
    lif_gemm_fused<<<grid, block, 0, stream>>>(x, W, bias, out);
}